// AtomLayer_54992761258849
// MI455X (gfx1250) — compile-verified
//
#include <hip/hip_runtime.h>

#ifndef __has_builtin
#define __has_builtin(x) 0
#endif

#define NB 64
#define NA 256
#define NT 2048
#define NK 16

typedef float v4f __attribute__((ext_vector_type(4)));

// ---------------------------------------------------------------------------
// Phase 1: one thread per (b,t).
//  - mask[b,a,t] = (x >= thr)                     (streaming NT store)
//  - bestVal[b,t], bestIdx[b,t] = max / argmax over A of thresholded x
//    (strict '>' keeps the FIRST occurrence, matching jnp.argmax)
// Consecutive threads = consecutive t = consecutive addresses -> coalesced;
// x and mask are touched exactly once -> non-temporal.
// ---------------------------------------------------------------------------
__global__ __launch_bounds__(256) void atom_phase1(
    const float* __restrict__ x, const float* __restrict__ thrp,
    float* __restrict__ maskOut,
    float* __restrict__ bestVal, int* __restrict__ bestIdx)
{
    const float thr = thrp[0];
    const int t = blockIdx.x * blockDim.x + threadIdx.x;   // 0..NT-1
    const int b = blockIdx.y;                              // 0..NB-1
    const size_t base = (size_t)b * NA * NT + (size_t)t;

    float bv = -1.0f;   // first iteration (tv >= 0) always updates -> argmax=0 if all zero
    int   ba = 0;
    #pragma unroll 4
    for (int a = 0; a < NA; ++a) {
        const size_t i = base + (size_t)a * NT;
        const float v = __builtin_nontemporal_load(&x[i]);
        const bool valid = (v >= thr);
        __builtin_nontemporal_store(valid ? 1.0f : 0.0f, &maskOut[i]);
        const float tv = valid ? v : 0.0f;
        if (tv > bv) { bv = tv; ba = a; }
    }
    bestVal[(size_t)b * NT + t] = bv;
    bestIdx[(size_t)b * NT + t] = ba;
}

// ---------------------------------------------------------------------------
// Phase 2: one block per row (b,a). Stage bestVal/bestIdx[b,:] (16 KB, L2-hot)
// into LDS via CDNA5 async global->LDS DMA, build the sparse candidate list
// with LDS atomics, select top-16 by 64-bit-key block max-reduction, fill the
// remaining slots with ascending zero-entries (jax stable-sort semantics),
// then emit the ENTIRE feat row (zeros + ones) as streaming b128 NT stores
// and write the indices.
// Key = (float_bits(val) << 32) | (NT-1-t): bigger value wins; equal value ->
// smaller t wins. val > 0 => key != 0, so 0 means "empty/consumed".
// ---------------------------------------------------------------------------
__global__ __launch_bounds__(256) void atom_phase2(
    const float* __restrict__ bestVal, const int* __restrict__ bestIdx,
    float* __restrict__ feat, float* __restrict__ idxOut)
{
    __shared__ __align__(16) float sVal[NT];
    __shared__ __align__(16) int   sIdx[NT];
    __shared__ unsigned long long  keys[NT];
    __shared__ unsigned char       isM[NT];     // matched flags, later reused as selected flags
    __shared__ unsigned long long  red[256];
    __shared__ int                 selT[NK];
    __shared__ int                 cnt;

    const int tid = threadIdx.x;
    const int row = blockIdx.x;          // b*NA + a
    const int b   = row >> 8;            // / NA
    const int a   = row & (NA - 1);      // % NA
    const float* gV = bestVal + (size_t)b * NT;
    const int*   gI = bestIdx + (size_t)b * NT;

    if (tid == 0) cnt = 0;

    // --- async DMA global -> LDS: 2 x b128 per array per lane (gfx1250 path) ---
    #pragma unroll
    for (int part = 0; part < 2; ++part) {
        const int e = part * 1024 + tid * 4;                   // 4 dwords per lane
        const unsigned ldsV = (unsigned)(size_t)&sVal[e];      // low 32b = LDS offset
        const unsigned long long gaV = (unsigned long long)(size_t)(gV + e);
        asm volatile("global_load_async_to_lds_b128 %0, %1, off"
                     :: "v"(ldsV), "v"(gaV) : "memory");
        const unsigned ldsI = (unsigned)(size_t)&sIdx[e];
        const unsigned long long gaI = (unsigned long long)(size_t)(gI + e);
        asm volatile("global_load_async_to_lds_b128 %0, %1, off"
                     :: "v"(ldsI), "v"(gaI) : "memory");
    }
#if __has_builtin(__builtin_amdgcn_s_wait_asynccnt)
    __builtin_amdgcn_s_wait_asynccnt(0);
#else
    asm volatile("s_wait_asynccnt 0x0" ::: "memory");
#endif
    __syncthreads();

    // --- append matched candidates (sparse: avg 8 per row) ---
    #pragma unroll
    for (int k = 0; k < NT / 256; ++k) {
        const int t = tid + k * 256;
        const float v  = sVal[t];
        const int   ai = sIdx[t];
        const bool  m  = (ai == a) && (v > 0.0f);
        isM[t] = m ? (unsigned char)1 : (unsigned char)0;
        if (m) {
            const int p = atomicAdd(&cnt, 1);
            keys[p] = ((unsigned long long)__float_as_uint(v) << 32)
                    | (unsigned long long)(unsigned)(NT - 1 - t);
        }
    }
    __syncthreads();
    const int M = cnt;

    // --- up to 16 rounds of block-wide max selection over keys ---
    int j = 0;
    for (; j < NK; ++j) {
        unsigned long long lm = 0ULL;
        for (int i = tid; i < M; i += 256) {
            const unsigned long long kk = keys[i];
            if (kk > lm) lm = kk;
        }
        red[tid] = lm;
        __syncthreads();
        #pragma unroll
        for (int s = 128; s > 0; s >>= 1) {
            if (tid < s) {
                const unsigned long long o = red[tid + s];
                if (o > red[tid]) red[tid] = o;
            }
            __syncthreads();
        }
        const unsigned long long best = red[0];   // uniform across block
        __syncthreads();
        if (best == 0ULL) break;                  // uniform break
        if (tid == 0) selT[j] = (NT - 1) - (int)(best & 0xFFFFFFFFULL);
        for (int i = tid; i < M; i += 256) {
            if (keys[i] == best) keys[i] = 0ULL;  // consume winner
        }
        __syncthreads();
    }

    // --- fill remaining slots with smallest-index zero entries ---
    if (tid == 0) {
        int c = j, t = 0;
        while (c < NK && t < NT) {
            if (!isM[t]) selT[c++] = t;
            ++t;
        }
    }
    __syncthreads();

    // --- reuse isM as "selected" flags ---
    #pragma unroll
    for (int k = 0; k < NT / 256; ++k) isM[tid + k * 256] = 0;
    __syncthreads();
    if (tid < NK) isM[selT[tid]] = 1;
    __syncthreads();

    // --- emit the whole feat row as streaming b128 NT stores ---
    float* featRow = feat + (size_t)row * NT;
    #pragma unroll
    for (int part = 0; part < 2; ++part) {
        const int e = part * 1024 + tid * 4;
        v4f o;
        o.x = isM[e + 0] ? 1.0f : 0.0f;
        o.y = isM[e + 1] ? 1.0f : 0.0f;
        o.z = isM[e + 2] ? 1.0f : 0.0f;
        o.w = isM[e + 3] ? 1.0f : 0.0f;
        __builtin_nontemporal_store(o, (v4f*)(featRow + e));
    }

    if (tid < NK) {
        idxOut[(size_t)row * NK + tid] = (float)selT[tid];
    }
}

// ---------------------------------------------------------------------------
extern "C" void kernel_launch(void* const* d_in, const int* in_sizes, int n_in,
                              void* d_out, int out_size, void* d_ws, size_t ws_size,
                              hipStream_t stream)
{
    (void)in_sizes; (void)n_in; (void)out_size; (void)ws_size;

    const float* x    = (const float*)d_in[0];
    const float* thrp = (const float*)d_in[1];

    // Outputs concatenated flat in return order: feat | indices | mask
    float* feat    = (float*)d_out;
    float* idxOut  = feat   + (size_t)NB * NA * NT;
    float* maskOut = idxOut + (size_t)NB * NA * NK;

    // Workspace: bestVal [NB*NT] f32, bestIdx [NB*NT] i32  (1 MiB total)
    float* bestVal = (float*)d_ws;
    int*   bestIdx = (int*)((char*)d_ws + sizeof(float) * (size_t)NB * NT);

    dim3 g1(NT / 256, NB);
    atom_phase1<<<g1, 256, 0, stream>>>(x, thrp, maskOut, bestVal, bestIdx);
    atom_phase2<<<NB * NA, 256, 0, stream>>>(bestVal, bestIdx, feat, idxOut);
}